// TFMambaBlock_81063212745234
// MI455X (gfx1250) — compile-verified
//
#include <hip/hip_runtime.h>
#include <hip/hip_bf16.h>

// ---------------------------------------------------------------------------
// TF-Mamba block for gfx1250 (MI455X).  Latency-bound workload (~8 GFLOP,
// ~5 MB activations): fuse each mamba_residual into one workgroup-per-sequence
// kernel with the whole sequence resident in LDS (CDNA5: 320KB/WGP), use
// V_WMMA_F32_16X16X4_F32 for every matmul, wave32 shfl-pair reduction for the
// selective-scan einsum.
// ---------------------------------------------------------------------------

typedef __attribute__((ext_vector_type(2))) float v2f;
typedef __attribute__((ext_vector_type(8))) float v8f;

#define CDIM 64
#define EDIM 128
#define NDIM 16
#define RDIM 4
#define BB   2
#define TT   128
#define FF   80

__device__ inline v8f wmma4(v2f a, v2f b, v8f c) {
  // D = A(16x4,f32) * B(4x16,f32) + C(16x16,f32)
  return __builtin_amdgcn_wmma_f32_16x16x4_f32(false, a, false, b, (short)0, c,
                                               false, false);
}

__device__ inline float sigmoidf_(float v) { return 1.0f / (1.0f + __expf(-v)); }
__device__ inline float siluf_(float v)    { return v * sigmoidf_(v); }
__device__ inline float softplusf_(float v) {
  return (v > 20.0f) ? v : log1pf(__expf(v));
}

// ---------------------------------------------------------------------------
// One workgroup = one sequence.  256 threads = 8 wave32.
// LDS layout: hbuf(L*64) | xa(L*128) | zb(L*128) | xb(L*128) | dbc(L*48)
// xa: raw inproj xin, later reused as y (scan output).  xb: conv+silu xin.
// ---------------------------------------------------------------------------
__global__ __launch_bounds__(256)
void mamba_seq_kernel(const float* __restrict__ xseq,      // (S, L, C)
                      const float* __restrict__ norm_w,    // (4, C)
                      const float* __restrict__ inproj_w,  // (4, 2*ED, C)
                      const float* __restrict__ conv_w,    // (4, ED, 4)
                      const float* __restrict__ conv_b,    // (4, ED)
                      const float* __restrict__ xproj_w,   // (4, 36, ED)
                      const float* __restrict__ dtproj_w,  // (4, ED, 4)
                      const float* __restrict__ dtproj_b,  // (4, ED)
                      const float* __restrict__ A_log,     // (4, ED, N)
                      const float* __restrict__ Dp,        // (4, ED)
                      const float* __restrict__ outproj_w, // (4, C, ED)
                      float* __restrict__ ycat,            // (S, L, 2C)
                      int L, int iParam, int reverse, int colOff)
{
  extern __shared__ float lds[];
  float* hbuf = lds;                  // L * 64
  float* xa   = hbuf + L * CDIM;      // L * 128
  float* zb   = xa + L * EDIM;        // L * 128
  float* xb   = zb + L * EDIM;        // L * 128
  float* dbc  = xb + L * EDIM;        // L * 48 (36 valid, padded)

  const int tid  = threadIdx.x;
  const int lane = tid & 31;
  const int wave = tid >> 5;
  const int seq  = blockIdx.x;
  const float* xrow = xseq + (size_t)seq * L * CDIM;

  // ---- phase 1: RMSNorm (2 threads per row, shfl pair-reduce) -------------
  {
    const float* Wn = norm_w + iParam * CDIM;
    const int half = tid & 1;
    for (int r = tid >> 1; r < L; r += 128) {
      const int tin = reverse ? (L - 1 - r) : r;
      const float* xr = xrow + (size_t)tin * CDIM;
      float ss = 0.0f;
      #pragma unroll
      for (int c = 0; c < 32; ++c) { float v = xr[half * 32 + c]; ss += v * v; }
      ss += __shfl_xor(ss, 1, 32);
      const float sc = rsqrtf(ss * (1.0f / 64.0f) + 1e-5f);
      #pragma unroll
      for (int c = 0; c < 32; ++c) {
        int cc = half * 32 + c;
        hbuf[r * CDIM + cc] = xr[cc] * sc * Wn[cc];
      }
    }
  }
  __syncthreads();

  // ---- phase 2: inproj  xz = h @ Win^T  (L,64)x(64,256) via WMMA ----------
  {
    const float* Win = inproj_w + (size_t)iParam * 2 * EDIM * CDIM;
    const int mT = L >> 4;
    const int tot = mT * 16;                  // 16 n-tiles of 256 cols
    const int klo = (lane >> 4) * 2;
    for (int tile = wave; tile < tot; tile += 8) {
      const int mt = tile >> 4, nt = tile & 15;
      const int m = mt * 16 + (lane & 15);
      const int n = nt * 16 + (lane & 15);
      v8f acc = {};
      for (int ks = 0; ks < 16; ++ks) {
        const int k = ks * 4 + klo;
        v2f a, b;
        a.x = hbuf[m * CDIM + k];  a.y = hbuf[m * CDIM + k + 1];
        b.x = Win[n * CDIM + k];   b.y = Win[n * CDIM + k + 1];
        acc = wmma4(a, b, acc);
      }
      const int mrow = mt * 16 + (lane >> 4) * 8;
      const int col  = nt * 16 + (lane & 15);
      float* dst = (col < EDIM) ? (xa + col) : (zb + (col - EDIM));
      #pragma unroll
      for (int r = 0; r < 8; ++r) dst[(mrow + r) * EDIM] = acc[r];
    }
  }
  __syncthreads();

  // ---- phase 3: causal depthwise conv (K=4) + SiLU: xa -> xb --------------
  {
    const float* Wc = conv_w + (size_t)iParam * EDIM * 4;
    const float* bc = conv_b + (size_t)iParam * EDIM;
    for (int idx = tid; idx < L * EDIM; idx += 256) {
      const int t = idx >> 7, d = idx & 127;
      float s = bc[d];
      if (t >= 3) s += xa[(t - 3) * EDIM + d] * Wc[d * 4 + 0];
      if (t >= 2) s += xa[(t - 2) * EDIM + d] * Wc[d * 4 + 1];
      if (t >= 1) s += xa[(t - 1) * EDIM + d] * Wc[d * 4 + 2];
      s += xa[t * EDIM + d] * Wc[d * 4 + 3];
      xb[idx] = siluf_(s);
    }
  }
  __syncthreads();

  // ---- phase 4: xproj  dbc = xb @ Wx^T  (L,128)x(128,36->48 pad) ----------
  {
    const float* Wx = xproj_w + (size_t)iParam * 36 * EDIM;
    const int mT = L >> 4;
    const int tot = mT * 3;
    const int klo = (lane >> 4) * 2;
    for (int tile = wave; tile < tot; tile += 8) {
      const int mt = tile / 3, nt = tile % 3;
      const int m = mt * 16 + (lane & 15);
      const int n = nt * 16 + (lane & 15);
      const bool nv = (n < 36);
      v8f acc = {};
      for (int ks = 0; ks < 32; ++ks) {
        const int k = ks * 4 + klo;
        v2f a, b;
        a.x = xb[m * EDIM + k];          a.y = xb[m * EDIM + k + 1];
        b.x = nv ? Wx[n * EDIM + k] : 0.0f;
        b.y = nv ? Wx[n * EDIM + k + 1] : 0.0f;
        acc = wmma4(a, b, acc);
      }
      const int mrow = mt * 16 + (lane >> 4) * 8;
      const int col  = nt * 16 + (lane & 15);
      #pragma unroll
      for (int r = 0; r < 8; ++r) dbc[(mrow + r) * 48 + col] = acc[r];
    }
  }
  __syncthreads();

  // ---- phase 5: selective scan.  2 threads per d; 8 states each in VGPRs. -
  // No barriers inside: pair-reduction is a single intra-wave shfl_xor.
  {
    const int d = tid >> 1, half = tid & 1, n0 = half * 8;
    const float* Al  = A_log + ((size_t)iParam * EDIM + d) * NDIM + n0;
    const float* Wdt = dtproj_w + ((size_t)iParam * EDIM + d) * RDIM;
    const float bdt = dtproj_b[iParam * EDIM + d];
    const float Dd  = Dp[iParam * EDIM + d];
    const float w0 = Wdt[0], w1 = Wdt[1], w2 = Wdt[2], w3 = Wdt[3];
    float Areg[8], hs[8];
    #pragma unroll
    for (int j = 0; j < 8; ++j) { Areg[j] = -__expf(Al[j]); hs[j] = 0.0f; }

    for (int t = 0; t < L; ++t) {
      const float* row = dbc + t * 48;
      const float del =
          softplusf_(row[0] * w0 + row[1] * w1 + row[2] * w2 + row[3] * w3 + bdt);
      const float xv = xb[t * EDIM + d];
      const float dx = del * xv;
      float acc = 0.0f;
      #pragma unroll
      for (int j = 0; j < 8; ++j) {
        const float dA = __expf(del * Areg[j]);
        hs[j] = dA * hs[j] + dx * row[4 + n0 + j];   // Bm
        acc += hs[j] * row[20 + n0 + j];             // Cm
      }
      acc += __shfl_xor(acc, 1, 32);
      if (half == 0) {
        const float zv = zb[t * EDIM + d];
        xa[t * EDIM + d] = (acc + Dd * xv) * siluf_(zv);  // reuse xa as y
      }
    }
  }
  __syncthreads();

  // ---- phase 6: outproj  out = y @ Wout^T + x  -> ycat[:, :, colOff..] ----
  {
    const float* Wo = outproj_w + (size_t)iParam * CDIM * EDIM;
    float* yrow = ycat + (size_t)seq * L * (2 * CDIM) + colOff;
    const int mT = L >> 4;
    const int tot = mT * 4;
    const int klo = (lane >> 4) * 2;
    for (int tile = wave; tile < tot; tile += 8) {
      const int mt = tile >> 2, nt = tile & 3;
      const int m = mt * 16 + (lane & 15);
      const int n = nt * 16 + (lane & 15);
      v8f acc = {};
      for (int ks = 0; ks < 32; ++ks) {
        const int k = ks * 4 + klo;
        v2f a, b;
        a.x = xa[m * EDIM + k];  a.y = xa[m * EDIM + k + 1];
        b.x = Wo[n * EDIM + k];  b.y = Wo[n * EDIM + k + 1];
        acc = wmma4(a, b, acc);
      }
      const int mrow = mt * 16 + (lane >> 4) * 8;
      const int col  = nt * 16 + (lane & 15);
      #pragma unroll
      for (int r = 0; r < 8; ++r) {
        const int t = mrow + r;
        const int tout = reverse ? (L - 1 - t) : t;
        yrow[(size_t)tout * (2 * CDIM) + col] =
            acc[r] + xrow[(size_t)tout * CDIM + col];
      }
    }
  }
}

// ---------------------------------------------------------------------------
// out = ycat(rows,128) @ W(128,64) + bias + resid(rows,64).  WMMA, 1 wave/mtile
// ---------------------------------------------------------------------------
__global__ __launch_bounds__(256)
void linres_kernel(const float* __restrict__ ycat, const float* __restrict__ W,
                   const float* __restrict__ bias, const float* __restrict__ resid,
                   float* __restrict__ out, int rows)
{
  const int lane = threadIdx.x & 31, wave = threadIdx.x >> 5;
  const int mt = blockIdx.x * 8 + wave;
  if (mt >= (rows >> 4)) return;          // wave-uniform
  const int m = mt * 16 + (lane & 15);
  const int klo = (lane >> 4) * 2;
  for (int nt = 0; nt < 4; ++nt) {
    const int n = nt * 16 + (lane & 15);
    v8f acc = {};
    for (int ks = 0; ks < 32; ++ks) {
      const int k = ks * 4 + klo;
      v2f a, b;
      a.x = ycat[(size_t)m * 128 + k];  a.y = ycat[(size_t)m * 128 + k + 1];
      b.x = W[(size_t)k * 64 + n];      b.y = W[(size_t)(k + 1) * 64 + n];
      acc = wmma4(a, b, acc);
    }
    const int mrow = mt * 16 + (lane >> 4) * 8;
    const int col  = nt * 16 + (lane & 15);
    const float bv = bias[col];
    #pragma unroll
    for (int r = 0; r < 8; ++r) {
      const size_t o = (size_t)(mrow + r) * 64 + col;
      out[o] = acc[r] + bv + resid[o];
    }
  }
}

// ---------------------------------------------------------------------------
// Layout shuffles
// ---------------------------------------------------------------------------
__global__ void t_in_kernel(const float* __restrict__ x, float* __restrict__ xt, int n) {
  int idx = blockIdx.x * 256 + threadIdx.x; if (idx >= n) return;
  int c = idx & 63; int r = idx >> 6; int t = r % TT; r /= TT;
  int f = r % FF; int b = r / FF;
  xt[idx] = x[(((size_t)b * CDIM + c) * TT + t) * FF + f];
}
__global__ void t_mid_kernel(const float* __restrict__ yt, float* __restrict__ yt2, int n) {
  int idx = blockIdx.x * 256 + threadIdx.x; if (idx >= n) return;
  int c = idx & 63; int r = idx >> 6; int f = r % FF; r /= FF;
  int t = r % TT; int b = r / TT;
  yt2[idx] = yt[(((size_t)b * FF + f) * TT + t) * CDIM + c];
}
__global__ void t_out_kernel(const float* __restrict__ yf, float* __restrict__ out, int n) {
  int idx = blockIdx.x * 256 + threadIdx.x; if (idx >= n) return;
  int f = idx % FF; int r = idx / FF; int t = r % TT; r /= TT;
  int c = r % CDIM; int b = r / CDIM;
  out[idx] = yf[(((size_t)b * TT + t) * FF + f) * CDIM + c];
}

// ---------------------------------------------------------------------------
extern "C" void kernel_launch(void* const* d_in, const int* in_sizes, int n_in,
                              void* d_out, int out_size, void* d_ws, size_t ws_size,
                              hipStream_t stream) {
  const float* x        = (const float*)d_in[0];
  const float* norm_w   = (const float*)d_in[1];
  const float* inproj_w = (const float*)d_in[2];
  const float* conv_w   = (const float*)d_in[3];
  const float* conv_b   = (const float*)d_in[4];
  const float* xproj_w  = (const float*)d_in[5];
  const float* dtproj_w = (const float*)d_in[6];
  const float* dtproj_b = (const float*)d_in[7];
  const float* A_log    = (const float*)d_in[8];
  const float* Dp       = (const float*)d_in[9];
  const float* outproj_w= (const float*)d_in[10];
  const float* tlin_w   = (const float*)d_in[11];
  const float* tlin_b   = (const float*)d_in[12];
  const float* flin_w   = (const float*)d_in[13];
  const float* flin_b   = (const float*)d_in[14];

  float* ws = (float*)d_ws;
  const int S1 = BB * FF, L1 = TT;        // 160 x 128
  const int S2 = BB * TT, L2 = FF;        // 256 x 80
  const int NE = BB * CDIM * TT * FF;     // 1,310,720 elements

  float* xt    = ws;                 size_t o = (size_t)S1 * L1 * CDIM;
  float* ycat1 = ws + o;             o += (size_t)S1 * L1 * 2 * CDIM;
  float* yt    = ws + o;             o += (size_t)S1 * L1 * CDIM;
  float* yt2   = ws + o;             o += (size_t)S2 * L2 * CDIM;
  float* ycat2 = ws + o;             o += (size_t)S2 * L2 * 2 * CDIM;
  float* yf    = ws + o;

  const int tb = (NE + 255) / 256;
  t_in_kernel<<<tb, 256, 0, stream>>>(x, xt, NE);

  const size_t lds1 = (size_t)(L1 * CDIM + 3 * L1 * EDIM + L1 * 48) * sizeof(float);
  mamba_seq_kernel<<<S1, 256, lds1, stream>>>(xt, norm_w, inproj_w, conv_w, conv_b,
      xproj_w, dtproj_w, dtproj_b, A_log, Dp, outproj_w, ycat1, L1, 0, 0, 0);
  mamba_seq_kernel<<<S1, 256, lds1, stream>>>(xt, norm_w, inproj_w, conv_w, conv_b,
      xproj_w, dtproj_w, dtproj_b, A_log, Dp, outproj_w, ycat1, L1, 1, 1, CDIM);
  linres_kernel<<<(S1 * L1 / 16) / 8, 256, 0, stream>>>(ycat1, tlin_w, tlin_b, xt, yt,
                                                        S1 * L1);

  t_mid_kernel<<<tb, 256, 0, stream>>>(yt, yt2, NE);

  const size_t lds2 = (size_t)(L2 * CDIM + 3 * L2 * EDIM + L2 * 48) * sizeof(float);
  mamba_seq_kernel<<<S2, 256, lds2, stream>>>(yt2, norm_w, inproj_w, conv_w, conv_b,
      xproj_w, dtproj_w, dtproj_b, A_log, Dp, outproj_w, ycat2, L2, 2, 0, 0);
  mamba_seq_kernel<<<S2, 256, lds2, stream>>>(yt2, norm_w, inproj_w, conv_w, conv_b,
      xproj_w, dtproj_w, dtproj_b, A_log, Dp, outproj_w, ycat2, L2, 3, 1, CDIM);
  linres_kernel<<<(S2 * L2 / 16) / 8, 256, 0, stream>>>(ycat2, flin_w, flin_b, yt2, yf,
                                                        S2 * L2);

  t_out_kernel<<<tb, 256, 0, stream>>>(yf, (float*)d_out, NE);
}